// MergedColumnParallelLinearWithLoRA_79800492359935
// MI455X (gfx1250) — compile-verified
//
#include <hip/hip_runtime.h>

// MergedColumnParallelLinearWithLoRA for MI455X (gfx1250, wave32, WMMA).
// out[T, 2I] = x @ W^T + (x @ A^T) @ B^T   (per-slice LoRA, rank 16)
// T=16384, H=2048, I=5632, N=11264. fp32 throughout via V_WMMA_F32_16X16X4_F32.
// HBM-bound (~0.96 GB traffic, ~41us floor at 23.3 TB/s); weight (92MB),
// A (256KB), B (720KB), la (2MB) all fit in the 192MB L2.

typedef __attribute__((ext_vector_type(2))) float v2f;
typedef __attribute__((ext_vector_type(4))) float v4f;
typedef __attribute__((ext_vector_type(8))) float v8f;

#define TOKENS 16384
#define HIDDEN 2048
#define INTER  5632
#define NCOLS  (2 * INTER)  // 11264
#define RANK   16

// ---------------------------------------------------------------------------
// Stage 1: la[t, s*16+r] = sum_h x[t,h] * A_buffer[s*16+r, h]
// One wave (32 threads) per 16-token tile; x A-fragment shared across both
// slices; K=2048 swept in steps of 4 with V_WMMA_F32_16X16X4_F32.
// A-matrix layout (32-bit, 16x4): lanes 0-15 -> M=lane, VGPR0/1 = K0/K1;
// lanes 16-31 -> M=lane-16, VGPR0/1 = K2/K3  => per-lane float2 at k0+klane.
// B-matrix (4x16) mirrors it: lane holds B[k0+klane+{0,1}][lane&15].
// ---------------------------------------------------------------------------
__global__ __launch_bounds__(32) void lora_a_kernel(
    const float* __restrict__ x, const float* __restrict__ A,
    float* __restrict__ la) {
  const int lane  = threadIdx.x;
  const int mr    = lane & 15;
  const int klane = (lane >> 4) * 2;
  const int m0    = blockIdx.x * 16;

  const float* xrow  = x + (size_t)(m0 + mr) * HIDDEN;
  const float* arow0 = A + (size_t)mr * HIDDEN;           // slice 0: rows 0..15
  const float* arow1 = A + (size_t)(16 + mr) * HIDDEN;    // slice 1: rows 16..31

  v8f acc0 = {}, acc1 = {};
  for (int k0 = 0; k0 < HIDDEN; k0 += 4) {
    v2f a  = *(const v2f*)(xrow  + k0 + klane);
    v2f b0 = *(const v2f*)(arow0 + k0 + klane);
    v2f b1 = *(const v2f*)(arow1 + k0 + klane);
    acc0 = __builtin_amdgcn_wmma_f32_16x16x4_f32(false, a, false, b0,
                                                 (short)0, acc0, false, false);
    acc1 = __builtin_amdgcn_wmma_f32_16x16x4_f32(false, a, false, b1,
                                                 (short)0, acc1, false, false);
  }
  // C/D layout: VGPR v -> row M=v (lanes 0-15) or M=v+8 (lanes 16-31), N=lane&15
  const int rowhi = (lane >> 4) * 8;
  for (int v = 0; v < 8; ++v) {
    const int row = m0 + v + rowhi;
    la[(size_t)row * (2 * RANK) + mr]        = acc0[v];
    la[(size_t)row * (2 * RANK) + RANK + mr] = acc1[v];
  }
}

// ---------------------------------------------------------------------------
// Stage 2: main GEMM + fused LoRA-B, 128x128 tile per 256-thread block.
// 8 waves = 2(M) x 4(N); each wave owns 4x2 tiles of 16x16 (64 accum VGPRs).
// K staged through LDS in BK=32 chunks, padded stride 36 floats (144B = 16B
// aligned rows, conflict-free b64 fragment reads), next chunk's global loads
// issued before the compute loop to overlap with WMMA.
// ---------------------------------------------------------------------------
#define BM  128
#define BN  128
#define BK  32
#define LDP 36  // padded row stride (floats): 144B rows -> 16B aligned, no bank conflicts

__global__ __launch_bounds__(256) void gemm_lora_kernel(
    const float* __restrict__ x, const float* __restrict__ w,
    const float* __restrict__ Bbuf, const float* __restrict__ la,
    float* __restrict__ out) {
  __shared__ float lX[BM * LDP];
  __shared__ float lW[BN * LDP];

  const int tid   = threadIdx.x;
  const int lane  = tid & 31;
  const int wave  = tid >> 5;
  const int waveM = wave >> 2;  // 0..1  -> 64 rows each
  const int waveN = wave & 3;   // 0..3  -> 32 cols each
  const int mr    = lane & 15;
  const int klane = (lane >> 4) * 2;
  const int m0    = blockIdx.y * BM;
  const int n0    = blockIdx.x * BN;

  v8f acc[4][2] = {};

  // Global->register staging: 128x32 floats for X and W each (8 float4/thread)
  v4f rx[4], rw[4];
#define LOAD_TILE(kb)                                                        \
  do {                                                                       \
    _Pragma("unroll") for (int i = 0; i < 4; ++i) {                          \
      const int q = tid + 256 * i;                                           \
      const int row = q >> 3, c4 = (q & 7) * 4;                              \
      rx[i] = *(const v4f*)(x + (size_t)(m0 + row) * HIDDEN + (kb) + c4);    \
      rw[i] = *(const v4f*)(w + (size_t)(n0 + row) * HIDDEN + (kb) + c4);    \
    }                                                                        \
  } while (0)

  LOAD_TILE(0);
  for (int kb = 0; kb < HIDDEN; kb += BK) {
    __syncthreads();
#pragma unroll
    for (int i = 0; i < 4; ++i) {
      const int q = tid + 256 * i;
      const int row = q >> 3, c4 = (q & 7) * 4;
      *(v4f*)(&lX[row * LDP + c4]) = rx[i];
      *(v4f*)(&lW[row * LDP + c4]) = rw[i];
    }
    __syncthreads();
    if (kb + BK < HIDDEN) LOAD_TILE(kb + BK);  // overlap next loads w/ compute

#pragma unroll
    for (int kk = 0; kk < BK; kk += 4) {
      v2f af[4], bf[2];
#pragma unroll
      for (int mt = 0; mt < 4; ++mt)
        af[mt] = *(const v2f*)(&lX[(waveM * 64 + mt * 16 + mr) * LDP + kk + klane]);
#pragma unroll
      for (int nt = 0; nt < 2; ++nt)
        bf[nt] = *(const v2f*)(&lW[(waveN * 32 + nt * 16 + mr) * LDP + kk + klane]);
#pragma unroll
      for (int mt = 0; mt < 4; ++mt)
#pragma unroll
        for (int nt = 0; nt < 2; ++nt)
          acc[mt][nt] = __builtin_amdgcn_wmma_f32_16x16x4_f32(
              false, af[mt], false, bf[nt], (short)0, acc[mt][nt], false, false);
    }
  }

  // Fused LoRA-B: acc += la_tile(16x16) @ B_tile^T, rank 16 = 4 WMMA K-steps.
  // INTER (5632) is a multiple of BN (128) -> whole block in one slice.
  const int s = (n0 >= INTER) ? 1 : 0;
#pragma unroll
  for (int rr = 0; rr < RANK; rr += 4) {
    v2f bfr[2];
#pragma unroll
    for (int nt = 0; nt < 2; ++nt) {
      const int n = n0 + waveN * 32 + nt * 16 + mr;
      bfr[nt] = *(const v2f*)(Bbuf + (size_t)n * RANK + rr + klane);
    }
#pragma unroll
    for (int mt = 0; mt < 4; ++mt) {
      const int row = m0 + waveM * 64 + mt * 16 + mr;
      v2f laf = *(const v2f*)(la + (size_t)row * (2 * RANK) + s * RANK + rr + klane);
#pragma unroll
      for (int nt = 0; nt < 2; ++nt)
        acc[mt][nt] = __builtin_amdgcn_wmma_f32_16x16x4_f32(
            false, laf, false, bfr[nt], (short)0, acc[mt][nt], false, false);
    }
  }

  // Store: VGPR v -> row v (lanes 0-15) / v+8 (lanes 16-31), col = lane&15
  const int rowhi = (lane >> 4) * 8;
#pragma unroll
  for (int mt = 0; mt < 4; ++mt)
#pragma unroll
    for (int nt = 0; nt < 2; ++nt) {
      const int ncol = n0 + waveN * 32 + nt * 16 + mr;
#pragma unroll
      for (int v = 0; v < 8; ++v) {
        const int row = m0 + waveM * 64 + mt * 16 + v + rowhi;
        out[(size_t)row * NCOLS + ncol] = acc[mt][nt][v];
      }
    }
}

extern "C" void kernel_launch(void* const* d_in, const int* in_sizes, int n_in,
                              void* d_out, int out_size, void* d_ws, size_t ws_size,
                              hipStream_t stream) {
  const float* x    = (const float*)d_in[0];  // [16384, 2048]
  const float* wgt  = (const float*)d_in[1];  // [11264, 2048]
  const float* Abuf = (const float*)d_in[2];  // [32, 2048]
  const float* Bbuf = (const float*)d_in[3];  // [11264, 16]
  float* out = (float*)d_out;                 // [16384, 11264]
  float* la  = (float*)d_ws;                  // [16384, 32] = 2 MB scratch

  lora_a_kernel<<<dim3(TOKENS / 16), dim3(32), 0, stream>>>(x, Abuf, la);
  gemm_lora_kernel<<<dim3(NCOLS / BN, TOKENS / BM), dim3(256), 0, stream>>>(
      x, wgt, Bbuf, la, out);
}